// Adapter_83442624626825
// MI455X (gfx1250) — compile-verified
//
#include <hip/hip_runtime.h>
#include <stdint.h>

// Problem dimensions (fixed by the reference: T,B,D,V = 1024,16,512,1000)
#define T_DIM 1024
#define B_DIM 16
#define D_DIM 512
#define V_DIM 1000

// -----------------------------------------------------------------------------
// Kernel 1: argmax over V per (t,b) row.
// One wave32 per row. The 1000-float row is staged global->LDS with the CDNA5
// async DMA path (global_load_async_to_lds_b128, tracked by ASYNCcnt), then
// reduced from LDS. 8 chunks of (32 lanes x 16B) = 1024 floats cover the row;
// out-of-row lanes are clamped to the row base and masked in the compare.
// -----------------------------------------------------------------------------
__global__ __launch_bounds__(256) void ctc_argmax_kernel(
    const float* __restrict__ logit, int* __restrict__ labels) {
  __shared__ __align__(16) float stage[8][1024];  // 8 waves * 4KB = 32KB LDS
  const int wave = threadIdx.x >> 5;
  const int lane = threadIdx.x & 31;
  const int row  = blockIdx.x * 8 + wave;  // row index = t*B + b
  const float* rowp = logit + (size_t)row * V_DIM;

  // Issue 8 async b128 loads per wave: memory -> LDS, no VGPR data path.
  for (int c = 0; c < 8; ++c) {
    int f4 = c * 32 + lane;            // float4 index within the row (0..255)
    int cl = (f4 < 250) ? f4 : 0;      // clamp tail lanes (row = 250 float4s)
    unsigned lds = (unsigned)(uintptr_t)&stage[wave][c * 128 + lane * 4];
    unsigned long long ga = (unsigned long long)(const char*)(rowp + (size_t)cl * 4);
    asm volatile("global_load_async_to_lds_b128 %0, %1, off"
                 :: "v"(lds), "v"(ga) : "memory");
  }
  asm volatile("s_wait_asynccnt 0" ::: "memory");  // own-wave LDS writes done

  float bv = -3.402823466e+38f;
  int   bi = 0x7fffffff;
  for (int c = 0; c < 8; ++c) {
    const float4 v = *(const float4*)&stage[wave][c * 128 + lane * 4];
    const int fl = c * 128 + lane * 4;
    const float e[4] = {v.x, v.y, v.z, v.w};
#pragma unroll
    for (int j = 0; j < 4; ++j) {
      const int idx = fl + j;
      if (idx < V_DIM && (e[j] > bv || (e[j] == bv && idx < bi))) {
        bv = e[j];
        bi = idx;
      }
    }
  }
  // wave32 reduction, first-occurrence tie-break (matches jnp.argmax)
#pragma unroll
  for (int off = 16; off > 0; off >>= 1) {
    const float ov = __shfl_xor(bv, off, 32);
    const int   oi = __shfl_xor(bi, off, 32);
    if (ov > bv || (ov == bv && oi < bi)) { bv = ov; bi = oi; }
  }
  if (lane == 0) {
    const int t = row / B_DIM;
    const int b = row % B_DIM;
    labels[b * T_DIM + t] = bi;  // [b][t] layout for the scan kernel
  }
}

// -----------------------------------------------------------------------------
// Kernel 2: per-batch run-length scan. One 1024-thread block per batch b.
//   change[t] = valid[t] && (label[t] != label[t-1])   (label[-1] := -1)
//   seg_id    = inclusive_scan(change) - 1             (ballot/popc, wave32)
//   counts    -> weights 1/run_length; segment start offsets; new_lengths;
//   new_padding written as 0.0/1.0 floats (second tuple output).
// -----------------------------------------------------------------------------
__global__ __launch_bounds__(1024) void ctc_segmeta_kernel(
    const int* __restrict__ labels, const unsigned char* __restrict__ padding,
    float* __restrict__ w, int* __restrict__ segStart,
    int* __restrict__ newLen, float* __restrict__ padOut) {
  __shared__ unsigned waveTot[32];
  __shared__ unsigned waveExc[32];
  __shared__ int cnt[T_DIM];
  __shared__ int totalS;

  const int b = blockIdx.x;
  const int t = threadIdx.x;
  cnt[t] = 0;

  const int  lab   = labels[b * T_DIM + t];
  const int  prev  = (t == 0) ? -1 : labels[b * T_DIM + t - 1];
  const bool valid = (padding[b * T_DIM + t] == 0);
  const bool change = valid && (lab != prev);

  const unsigned mask = (unsigned)__ballot(change);  // wave32: low 32 bits
  const int lane = t & 31;
  const int wv   = t >> 5;
  const unsigned inclWave = __popc(mask & (0xffffffffu >> (31 - lane)));
  if (lane == 0) waveTot[wv] = __popc(mask);
  __syncthreads();

  if (t < 32) {  // first wave scans the 32 wave totals
    const unsigned v = waveTot[t];
    unsigned inc = v;
#pragma unroll
    for (int d = 1; d < 32; d <<= 1) {
      const unsigned n = __shfl_up(inc, d, 32);
      if (t >= d) inc += n;
    }
    waveExc[t] = inc - v;
    if (t == 31) totalS = (int)inc;
  }
  __syncthreads();

  const int incl = (int)(waveExc[wv] + inclWave);  // #changes in [0..t]
  const int seg  = incl - 1;                       // this timestep's segment
  if (change) segStart[b * T_DIM + seg] = t;
  if (valid && seg >= 0) atomicAdd(&cnt[seg], 1);
  __syncthreads();

  w[b * T_DIM + t] = (valid && seg >= 0) ? 1.0f / (float)cnt[seg] : 0.0f;
  if (t == 0) newLen[b] = totalS;
  // new_padding (B,T): True (1.0) where t >= new_length[b]
  padOut[b * T_DIM + t] = (t >= totalS) ? 1.0f : 0.0f;
}

// -----------------------------------------------------------------------------
// Kernel 3: segment-mean gather. One 128-thread block per (segment s, batch b);
// each thread owns one float4 of D=512. Runs are contiguous in t, so the
// segment is [t0, t1); invalid timesteps inside carry weight 0. Segments
// beyond new_length[b] are zero-filled (d_out is poisoned by the harness).
// Output layout (T',B,D): out[((s*B)+b)*D + d].
// -----------------------------------------------------------------------------
__global__ __launch_bounds__(128) void ctc_segavg_kernel(
    const float* __restrict__ rep, const float* __restrict__ w,
    const int* __restrict__ segStart, const int* __restrict__ newLen,
    float* __restrict__ out) {
  const int s = blockIdx.x;
  const int b = blockIdx.y;
  const int nl = newLen[b];
  float4 acc = make_float4(0.f, 0.f, 0.f, 0.f);
  if (s < nl) {
    const int t0 = segStart[b * T_DIM + s];
    const int t1 = (s + 1 < nl) ? segStart[b * T_DIM + s + 1] : T_DIM;
    for (int t = t0; t < t1; ++t) {
      const float ww = w[b * T_DIM + t];  // uniform scalar per t
      if (ww != 0.0f) {
        const float4 v =
            ((const float4*)(rep + ((size_t)t * B_DIM + b) * D_DIM))[threadIdx.x];
        acc.x += ww * v.x;
        acc.y += ww * v.y;
        acc.z += ww * v.z;
        acc.w += ww * v.w;
      }
    }
  }
  ((float4*)(out + ((size_t)s * B_DIM + b) * D_DIM))[threadIdx.x] = acc;
}

// -----------------------------------------------------------------------------
// Launch. Inputs (setup_inputs order): representation f32 (T,B,D),
// logit f32 (T,B,V), padding bool (B,T). Output: (T,B,D) f32 then (B,T)
// new_padding flattened after it (as 0.0/1.0 floats).
// Workspace layout: labels[int B*T] | segStart[int B*T] | w[f32 B*T] | newLen[int B]
// -----------------------------------------------------------------------------
extern "C" void kernel_launch(void* const* d_in, const int* in_sizes, int n_in,
                              void* d_out, int out_size, void* d_ws, size_t ws_size,
                              hipStream_t stream) {
  (void)in_sizes; (void)n_in; (void)out_size; (void)ws_size;
  const float* rep = (const float*)d_in[0];
  const float* logit = (const float*)d_in[1];
  const unsigned char* padding = (const unsigned char*)d_in[2];
  float* out = (float*)d_out;
  float* padOut = out + (size_t)T_DIM * B_DIM * D_DIM;

  int* labels = (int*)d_ws;
  int* segStart = labels + T_DIM * B_DIM;
  float* w = (float*)(segStart + T_DIM * B_DIM);
  int* newLen = (int*)(w + T_DIM * B_DIM);

  // 1) argmax over V: 16384 rows, 8 waves (rows) per 256-thread block
  ctc_argmax_kernel<<<(T_DIM * B_DIM) / 8, 256, 0, stream>>>(logit, labels);
  // 2) per-batch run-length scan + weights + new_padding
  ctc_segmeta_kernel<<<B_DIM, T_DIM, 0, stream>>>(labels, padding, w, segStart,
                                                  newLen, padOut);
  // 3) segment-mean gather into (T',B,D), zero-fill past new_length
  ctc_segavg_kernel<<<dim3(T_DIM, B_DIM), D_DIM / 4, 0, stream>>>(
      rep, w, segStart, newLen, out);
}